// DipoleNet_65412351918473
// MI455X (gfx1250) — compile-verified
//
#include <hip/hip_runtime.h>

typedef __attribute__((ext_vector_type(2))) float v2f;
typedef __attribute__((ext_vector_type(8))) float v8f;

#define HID 19

// ---------------------------------------------------------------------------
// Dense per-node linear: out[n, 0..19) = act( [xin | agg][n,:] @ W + b (+res) )
// One wave32 handles a 16-node tile using V_WMMA_F32_16X16X4_F32 (exact f32).
//   A-frag (16x4):  lane<16 holds row=lane, K=kb+0/kb+1 ; lane>=16 K=kb+2/kb+3
//   B-frag (4x16):  same K split, col = lane&15 (per N-tile)
//   C/D (16x16):    VGPR v: lanes0-15 -> M=v, lanes16-31 -> M=v+8, N=lane&15
// Weights/bias staged zero-padded in LDS -> all inner-loop loads are
// unconditional; K/N padding is exactly zero through the WMMA, so A may load
// don't-care bytes for k >= K (ws buffers carry tail slack for this).
// `out` must be a padded workspace buffer: stores need no row
// guard (full 16-row tiles always in-bounds).
// ---------------------------------------------------------------------------
template <int XD, int AD, bool RES, bool RELU>
__global__ __launch_bounds__(256) void linear_wmma_kernel(
    const float* __restrict__ xin, const float* __restrict__ agg,
    const float* __restrict__ W, const float* __restrict__ Bv,
    float* __restrict__ out, int N)
{
    constexpr int K  = XD + AD;
    constexpr int KP = (K + 3) & ~3;

    // ---- stage zero-padded W (KP x 32) and bias (32) in LDS ----
    __shared__ float Wlds[KP * 32];
    __shared__ float Blds[32];
    for (int i = threadIdx.x; i < KP * 32; i += blockDim.x) {
        const int k = i >> 5, c = i & 31;
        Wlds[i] = (k < K && c < HID) ? W[k * HID + c] : 0.0f;
    }
    if (threadIdx.x < 32)
        Blds[threadIdx.x] = (threadIdx.x < HID) ? Bv[threadIdx.x] : 0.0f;
    __syncthreads();

    const int lane = threadIdx.x & 31;
    const int wave = blockIdx.x * (blockDim.x >> 5) + (threadIdx.x >> 5);
    const int base = wave * 16;
    if (base >= N) return;                 // wave-uniform: EXEC stays all-ones

    const int hi  = lane >> 4;             // 0 or 1 (half-wave)
    const int l15 = lane & 15;
    int row = base + l15;
    if (row > N - 1) row = N - 1;          // clamp A-loads (layer-1 real input)

    const float* xr = xin + (size_t)row * XD;
    const float* ar = (AD > 0) ? (agg + (size_t)row * HID) : xr;

    // per-lane LDS base: element (k0, col) = Wlds[(kb + 2*hi)*32 + col]
    const int wb = hi * 64 + l15;          // += kb*32 ; +32 for k1 ; +16 tile1

    v8f acc0, acc1;
    {
        const float b0 = Blds[l15];
        const float b1 = Blds[16 + l15];
#pragma unroll
        for (int i = 0; i < 8; ++i) { acc0[i] = b0; acc1[i] = b1; }
    }

#pragma unroll
    for (int kb = 0; kb < KP; kb += 4) {
        const int k0 = kb + 2 * hi;
        const int k1 = k0 + 1;

        // branchless A loads: pointer select only; k >= K reads tail slack
        // whose product with the guaranteed-zero B column is 0.
        v2f a;
        if (AD > 0) {
            const float* p0 = (k0 < XD) ? (xr + k0) : (ar + (k0 - XD));
            const float* p1 = (k1 < XD) ? (xr + k1) : (ar + (k1 - XD));
            a[0] = *p0;
            a[1] = *p1;
        } else {
            a[0] = xr[(k0 < XD) ? k0 : 0];
            a[1] = xr[(k1 < XD) ? k1 : 0];
        }

        v2f bA, bB;
        bA[0] = Wlds[kb * 32 + wb];        // (k0, l15)
        bA[1] = Wlds[kb * 32 + wb + 32];   // (k1, l15)
        bB[0] = Wlds[kb * 32 + wb + 16];   // (k0, 16+l15)
        bB[1] = Wlds[kb * 32 + wb + 48];   // (k1, 16+l15)

        acc0 = __builtin_amdgcn_wmma_f32_16x16x4_f32(false, a, false, bA,
                                                     (short)0, acc0, false, false);
        acc1 = __builtin_amdgcn_wmma_f32_16x16x4_f32(false, a, false, bB,
                                                     (short)0, acc1, false, false);
    }

    // ---- epilogue: out buffer is padded to full tiles -> no row guards ----
    const size_t rbase = (size_t)(base + 8 * hi);
    float*       op = out + rbase * HID + l15;               // + v*HID
    const float* rp = RES ? (xin + rbase * XD + l15) : nullptr;

#pragma unroll
    for (int v = 0; v < 8; ++v) {
        float o0 = acc0[v];
        if (RES)  o0 += rp[v * XD];
        if (RELU) o0 = fmaxf(o0, 0.0f);
        op[v * HID] = o0;                   // immediate offset v*76B
    }
    if (16 + l15 < HID) {                   // single divergence for 8 stores
#pragma unroll
        for (int v = 0; v < 8; ++v) {
            float o1 = acc1[v];
            if (RES)  o1 += rp[v * XD + 16];
            if (RELU) o1 = fmaxf(o1, 0.0f);
            op[v * HID + 16] = o1;
        }
    }
}

// ---------------------------------------------------------------------------
// Edge scatter-add: agg[dst[e], c] += h[src[e], c]
// One thread per (edge, channel). 19 consecutive threads share one edge ->
// coalesced row reads; atomic adds land on contiguous 19-float rows in L2.
// h (38MB) and agg (38MB) both fit in the 192MB L2.
// ---------------------------------------------------------------------------
__global__ __launch_bounds__(256) void scatter_add_kernel(
    const float* __restrict__ h, const int* __restrict__ src,
    const int* __restrict__ dst, float* __restrict__ agg, int nEdges)
{
    long long t = (long long)blockIdx.x * blockDim.x + threadIdx.x;
    long long total = (long long)nEdges * HID;
    if (t >= total) return;
    int e = (int)(t / HID);
    int c = (int)(t - (long long)e * HID);
    float val = h[(size_t)src[e] * HID + c];
    atomicAdd(agg + (size_t)dst[e] * HID + c, val);
}

// ---------------------------------------------------------------------------
// Readout: pos[g] = (#nodes with batch <= g) - 1  (== cumsum(bincount)-1,
// handles empty graphs), then 19->10 relu -> 1 relu MLP.
// ---------------------------------------------------------------------------
__global__ __launch_bounds__(256) void readout_kernel(
    const float* __restrict__ h, const int* __restrict__ batch,
    int nNodes, int nGraphs,
    const float* __restrict__ wfc0, const float* __restrict__ bfc0,
    const float* __restrict__ wfc1, const float* __restrict__ bfc1,
    float* __restrict__ out)
{
    int g = blockIdx.x * blockDim.x + threadIdx.x;
    if (g >= nGraphs) return;
    int lo = 0, hi = nNodes;
    while (lo < hi) {
        int mid = (lo + hi) >> 1;
        if (batch[mid] <= g) lo = mid + 1; else hi = mid;
    }
    int pos = lo - 1;
    if (pos < 0) pos += nNodes;            // mimic JAX negative-index wrap
    const float* hr = h + (size_t)pos * HID;

    float hv[HID];
#pragma unroll
    for (int k = 0; k < HID; ++k) hv[k] = hr[k];

    float o = bfc1[0];
#pragma unroll
    for (int j = 0; j < 10; ++j) {
        float s = bfc0[j];
#pragma unroll
        for (int k = 0; k < HID; ++k) s += hv[k] * wfc0[k * 10 + j];
        s = fmaxf(s, 0.0f);
        o += s * wfc1[j];
    }
    out[g] = fmaxf(o, 0.0f);
}

// ---------------------------------------------------------------------------
extern "C" void kernel_launch(void* const* d_in, const int* in_sizes, int n_in,
                              void* d_out, int out_size, void* d_ws, size_t ws_size,
                              hipStream_t stream)
{
    const float* x     = (const float*)d_in[0];
    const int*   ei    = (const int*)d_in[1];
    const int*   batch = (const int*)d_in[2];
    const float* w0_1 = (const float*)d_in[3];  const float* b0_1 = (const float*)d_in[4];
    const float* w1_1 = (const float*)d_in[5];  const float* b1_1 = (const float*)d_in[6];
    const float* w0_2 = (const float*)d_in[7];  const float* b0_2 = (const float*)d_in[8];
    const float* w1_2 = (const float*)d_in[9];  const float* b1_2 = (const float*)d_in[10];
    const float* w0_3 = (const float*)d_in[11]; const float* b0_3 = (const float*)d_in[12];
    const float* w1_3 = (const float*)d_in[13]; const float* b1_3 = (const float*)d_in[14];
    const float* wfc0 = (const float*)d_in[15]; const float* bfc0 = (const float*)d_in[16];
    const float* wfc1 = (const float*)d_in[17]; const float* bfc1 = (const float*)d_in[18];

    const int N = in_sizes[0] / 11;
    const int E = in_sizes[1] / 2;
    const int G = out_size;
    const int* srcIdx = ei;        // edge_index[0]
    const int* dstIdx = ei + E;    // edge_index[1]

    // ws buffers padded to full 16-row tiles + 256B tail slack (over-read of
    // K-pad lanes and unguarded full-tile stores both stay in-bounds).
    const size_t Np  = ((size_t)N + 15) & ~(size_t)15;
    size_t per = ((Np * HID * sizeof(float)) + 256 + 255) & ~(size_t)255;
    char* ws = (char*)d_ws;
    float* hbuf = (float*)(ws);
    float* aggb = (float*)(ws + per);
    float* X1   = (float*)(ws + 2 * per);
    float* X2   = (float*)(ws + 3 * per);

    const int tiles = (N + 15) / 16;            // one wave per 16-node tile
    dim3 blkL(256);                              // 8 waves / block
    dim3 grdL((tiles + 7) / 8);
    long long totS = (long long)E * HID;
    dim3 blkS(256);
    dim3 grdS((unsigned)((totS + 255) / 256));
    size_t aggBytes = Np * HID * sizeof(float);

    // ---- Layer 1: in=11 -> 19 ----
    linear_wmma_kernel<11, 0,  false, false><<<grdL, blkL, 0, stream>>>(x,  nullptr, w0_1, b0_1, hbuf, N);
    hipMemsetAsync(aggb, 0, aggBytes, stream);
    scatter_add_kernel<<<grdS, blkS, 0, stream>>>(hbuf, srcIdx, dstIdx, aggb, E);
    linear_wmma_kernel<11, 19, false, true ><<<grdL, blkL, 0, stream>>>(x,  aggb,    w1_1, b1_1, X1,   N);

    // ---- Layer 2: 19 -> 19, residual ----
    linear_wmma_kernel<19, 0,  false, false><<<grdL, blkL, 0, stream>>>(X1, nullptr, w0_2, b0_2, hbuf, N);
    hipMemsetAsync(aggb, 0, aggBytes, stream);
    scatter_add_kernel<<<grdS, blkS, 0, stream>>>(hbuf, srcIdx, dstIdx, aggb, E);
    linear_wmma_kernel<19, 19, true,  true ><<<grdL, blkL, 0, stream>>>(X1, aggb,    w1_2, b1_2, X2,   N);

    // ---- Layer 3: 19 -> 19, residual ----
    linear_wmma_kernel<19, 0,  false, false><<<grdL, blkL, 0, stream>>>(X2, nullptr, w0_3, b0_3, hbuf, N);
    hipMemsetAsync(aggb, 0, aggBytes, stream);
    scatter_add_kernel<<<grdS, blkS, 0, stream>>>(hbuf, srcIdx, dstIdx, aggb, E);
    linear_wmma_kernel<19, 19, true,  true ><<<grdL, blkL, 0, stream>>>(X2, aggb,    w1_3, b1_3, X1,   N);

    // ---- Readout ----
    readout_kernel<<<dim3((G + 255) / 256), dim3(256), 0, stream>>>(
        X1, batch, N, G, wfc0, bfc0, wfc1, bfc1, (float*)d_out);

    (void)n_in; (void)ws_size;
}